// VmfProductPosterior_10952166605106
// MI455X (gfx1250) — compile-verified
//
#include <hip/hip_runtime.h>
#include <hip/hip_bf16.h>
#include <stdint.h>
#include <math.h>

// ---------------------------------------------------------------------------
// vMF product posterior forward for MI455X (gfx1250, wave32).
//   inputs : kappa_mu [B=2048, V=64, m=5] f32,  n_samples (=16)
//   outputs: samples  [B, 16, 320] f32  ++  log_prob [B, 16] f32 (concat flat)
//
// Store-bandwidth bound (~42MB out -> ~1.8us floor @ 23.3 TB/s). One block
// per b. CDNA5 paths:
//   * TDM tensor_load_to_lds   : stage kappa_mu[b] (1.25KB) into LDS
//   * TDM tensor_store_from_lds: stream the staged 20KB samples tile to HBM
//     (per-element 20B writes land in LDS, bank-conflict-free; the DMA gives
//      a clean wide store stream and frees the vector memory pipes)
//   * v_wmma_f32_16x16x4_f32 (B = ones): exact f32 cross-lane reducer for the
//     per-sample log-prob sum over V=64 (4 chained WMMAs).
// ---------------------------------------------------------------------------

typedef __attribute__((ext_vector_type(2))) float        v2f;
typedef __attribute__((ext_vector_type(8))) float        v8f;
typedef __attribute__((ext_vector_type(4))) unsigned int v4u;
typedef __attribute__((ext_vector_type(4))) int          v4i;
typedef __attribute__((ext_vector_type(8))) int          v8i;

#define B_V_DIM   64
#define B_M_DIM   5
#define N_SAMP    16
#define K_PROP    16
#define ZROW      (B_V_DIM * B_M_DIM)          /* 320 */
#define TILE_F    (N_SAMP * ZROW)              /* 5120 f32 = 20KB */
#define EPS_F     1e-5f
#define LOG2PI_F  1.8378770664093453f
#define LOG4X4_F  5.545177444479562f   /* 4*log(4) = (m-1)*log(m-1) */

#if __has_builtin(__builtin_amdgcn_tensor_load_to_lds) && \
    __has_builtin(__builtin_amdgcn_tensor_store_from_lds)
#define HAVE_TDM 1
#else
#define HAVE_TDM 0
#endif

// 1-D contiguous tensor DMA descriptor helper: nelem f32 global<->LDS.
#if HAVE_TDM
__device__ __forceinline__ void tdm_desc_1d(unsigned long long gaddr, unsigned ldsaddr,
                                            int nelem, v4u& g0, v8i& g1) {
  // group0: count=1 | lds_addr | global_addr[31:0] | global_addr[56:32] + type=2
  g0.x = 1u;
  g0.y = ldsaddr;
  g0.z = (unsigned)gaddr;
  g0.w = ((unsigned)(gaddr >> 32) & 0x01FFFFFFu) | 0x80000000u;
  // group1: data_size=4B; tensor_dim0=nelem; tensor_dim1=1; tile_dim0=nelem;
  //         tile_dim1/2=0 (unused); tensor_dim0_stride=nelem  (nelem < 65536)
  g1[0] = 0x20000;          // data_size = 2 (4 bytes)
  g1[1] = nelem << 16;      // tensor_dim0[15:0] in bits 63:48
  g1[2] = 1 << 16;          // tensor_dim1 = 1 in bits 95:80
  g1[3] = nelem << 16;      // tile_dim0 in bits 127:112
  g1[4] = 0;                // tile_dim1/2 unused
  g1[5] = nelem;            // tensor_dim0_stride[31:0]
  g1[6] = 0;
  g1[7] = 0;
}
#endif

// Deterministic counter-based RNG: pcg4d (Jarzynski & Olano).
__device__ __forceinline__ v4u pcg4d(v4u v) {
  v = v * 1664525u + 1013904223u;
  v.x += v.y * v.w; v.y += v.z * v.x; v.z += v.x * v.y; v.w += v.y * v.z;
  v ^= v >> 16;
  v.x += v.y * v.w; v.y += v.z * v.x; v.z += v.x * v.y; v.w += v.y * v.z;
  return v;
}

__device__ __forceinline__ float u01(unsigned r) {
  // uniform in (0,1): safe for log()
  return (float)(r >> 8) * 0x1p-24f + 0x1p-25f;
}

__device__ __forceinline__ float med3(float a, float b, float c) {
  // median of 3 uniforms ~ Beta(2,2)  (order-statistics identity; con=(m-1)/2=2)
  return a + b + c - fminf(a, fminf(b, c)) - fmaxf(a, fmaxf(b, c));
}

__global__ __launch_bounds__(256, 4)
void vmf_product_posterior_kernel(const float* __restrict__ kmu,
                                  float* __restrict__ out_samples,
                                  float* __restrict__ out_logp) {
  __shared__ float s_tile[ZROW];                        // kappa_mu[b] : 320 f32
  __shared__ float s_sc[B_V_DIM], s_kb[B_V_DIM], s_ka[B_V_DIM],
                   s_kd[B_V_DIM], s_ln[B_V_DIM];
  __shared__ float s_loc[B_V_DIM][B_M_DIM];
  __shared__ float s_hu[B_V_DIM][B_M_DIM];
  __shared__ float s_red[N_SAMP * 16];                  // per-(n, vlane) partials
  __shared__ __align__(16) float s_out[TILE_F];         // staged samples tile 20KB

  const int tid = threadIdx.x;
  const int b   = blockIdx.x;

  // ---- stage kappa_mu[b] (1.25KB) into LDS via the Tensor Data Mover -------
#if HAVE_TDM
  if (tid < 32) {  // one wave issues the DMA (EXEC ignored by TDM)
    unsigned long long g =
        (unsigned long long)(uintptr_t)(const void*)(kmu + (size_t)b * ZROW);
    v4u g0; v8i g1;
    tdm_desc_1d(g, (unsigned)(uintptr_t)(void*)&s_tile[0], ZROW, g0, g1);
    v4i gz = { 0, 0, 0, 0 };
#if defined(__clang_major__) && __clang_major__ >= 23
    v8i gz8 = { 0, 0, 0, 0, 0, 0, 0, 0 };
    __builtin_amdgcn_tensor_load_to_lds(g0, g1, gz, gz, gz8, 0);
#else
    __builtin_amdgcn_tensor_load_to_lds(g0, g1, gz, gz, 0);
#endif
    __builtin_amdgcn_s_wait_tensorcnt(0);
  }
#else
  for (int i = tid; i < ZROW; i += 256)
    s_tile[i] = kmu[(size_t)b * ZROW + i];
#endif
  __syncthreads();

  // ---- phase 1: per-v vMF constants (computed once, shared by 16 samples) --
  if (tid < B_V_DIM) {
    const int v = tid;
    float k0 = s_tile[v * 5 + 0], k1 = s_tile[v * 5 + 1], k2 = s_tile[v * 5 + 2],
          k3 = s_tile[v * 5 + 3], k4 = s_tile[v * 5 + 4];
    float sc  = sqrtf(k0 * k0 + k1 * k1 + k2 * k2 + k3 * k3 + k4 * k4);
    float inv = 1.0f / (sc + EPS_F);
    float l0 = k0 * inv, l1 = k1 * inv, l2 = k2 * inv, l3 = k3 * inv, l4 = k4 * inv;

    // Wood's constants (m=5): c, b (smooth Taylor switch), a, d
    float c  = sqrtf(4.0f * sc * sc + 16.0f);
    float bt = (c - 2.0f * sc) * 0.25f;
    float ba = 1.0f / sc;                              // (m-1)/(4*sc)
    float sw = fminf(fmaxf(sc - 10.0f, 0.0f), 1.0f);
    float kb = ba * sw + bt * (1.0f - sw);
    float ka = (4.0f + 2.0f * sc + c) * 0.25f;
    float kd = (4.0f * ka * kb) / (1.0f + kb) - LOG4X4_F;

    // log-normalizer: -((m/2-1)log(sc) - (m/2)log(2pi) - (sc + log_ive_1p5(sc)))
    float em2 = __expf(-2.0f * sc);
    float ive = sqrtf(2.0f / (3.14159265358979f * sc)) *
                ((1.0f + em2) * 0.5f - (1.0f - em2) / (2.0f * sc));
    float ln  = -(1.5f * __logf(sc) - 2.5f * LOG2PI_F - (sc + __logf(ive)));

    // Householder u = normalize(e1 - loc)  (+1e-5 in denominator per reference)
    float u0 = 1.0f - l0, u1 = -l1, u2 = -l2, u3 = -l3, u4 = -l4;
    float iu = 1.0f / (sqrtf(u0 * u0 + u1 * u1 + u2 * u2 + u3 * u3 + u4 * u4) + 1e-5f);

    s_sc[v] = sc; s_kb[v] = kb; s_ka[v] = ka; s_kd[v] = kd; s_ln[v] = ln;
    s_loc[v][0] = l0; s_loc[v][1] = l1; s_loc[v][2] = l2; s_loc[v][3] = l3; s_loc[v][4] = l4;
    s_hu[v][0] = u0 * iu; s_hu[v][1] = u1 * iu; s_hu[v][2] = u2 * iu;
    s_hu[v][3] = u3 * iu; s_hu[v][4] = u4 * iu;
  }
  __syncthreads();

  // ---- phase 2: sample + Householder; results into LDS staging tile --------
  const int n  = tid >> 4;    // sample index 0..15
  const int vb = tid & 15;    // v lane: 5-float stride, conflict-free (gcd(5,64)=1)
  float lp = 0.0f;

#pragma unroll
  for (int j = 0; j < 4; ++j) {
    const int v = vb + 16 * j;
    const float sc = s_sc[v], kb = s_kb[v], ka = s_ka[v], kd = s_kd[v], ln = s_ln[v];

    // Wood's rejection sampler: first accepted of K_PROP, else proposal 0
    float w = 0.0f, w0 = 0.0f;
    bool done = false;
    for (int k = 0; k < K_PROP; ++k) {
      v4u ctr = { (unsigned)b, (unsigned)((n << 8) | v), (unsigned)k, 0x9E3779B9u };
      v4u r   = pcg4d(ctr);
      float e   = med3(u01(r.x), u01(r.y), u01(r.z));   // Beta(2,2)
      float uu  = u01(r.w);
      float den = 1.0f - (1.0f - kb) * e;
      float wp  = (1.0f - (1.0f + kb) * e) / den;
      float t   = 2.0f * ka * kb / den;
      if (k == 0) w0 = wp;
      if (4.0f * __logf(t) - t + kd > __logf(uu)) { w = wp; done = true; break; }
    }
    if (!done) w = w0;

    // tangent Gaussian (Box-Muller), normalize, lift to sphere
    v4u ctr = { (unsigned)b, (unsigned)((n << 8) | v), 16u, 0x9E3779B9u };
    v4u r   = pcg4d(ctr);
    float a1 = u01(r.x), a2 = u01(r.y), a3 = u01(r.z), a4 = u01(r.w);
    float rd0 = sqrtf(-2.0f * __logf(a1));
    float rd1 = sqrtf(-2.0f * __logf(a3));
    float sn0, cs0, sn1, cs1;
    __sincosf(6.2831853071795865f * a2, &sn0, &cs0);
    __sincosf(6.2831853071795865f * a4, &sn1, &cs1);
    float g0 = rd0 * cs0, g1 = rd0 * sn0, g2 = rd1 * cs1, g3 = rd1 * sn1;
    float gi = rsqrtf(g0 * g0 + g1 * g1 + g2 * g2 + g3 * g3);
    float wt = sqrtf(fmaxf(1.0f - w * w, 1e-10f));
    float x0 = w, x1 = wt * g0 * gi, x2 = wt * g1 * gi,
          x3 = wt * g2 * gi, x4 = wt * g3 * gi;

    // Householder reflect e1 -> loc
    float h0 = s_hu[v][0], h1 = s_hu[v][1], h2 = s_hu[v][2],
          h3 = s_hu[v][3], h4 = s_hu[v][4];
    float d2 = 2.0f * (x0 * h0 + x1 * h1 + x2 * h2 + x3 * h3 + x4 * h4);
    float z0 = x0 - d2 * h0, z1 = x1 - d2 * h1, z2 = x2 - d2 * h2,
          z3 = x3 - d2 * h3, z4 = x4 - d2 * h4;

    const int sbase = n * ZROW + v * B_M_DIM;
    s_out[sbase + 0] = z0; s_out[sbase + 1] = z1; s_out[sbase + 2] = z2;
    s_out[sbase + 3] = z3; s_out[sbase + 4] = z4;

    lp += (s_loc[v][0] * z0 + s_loc[v][1] * z1 + s_loc[v][2] * z2 +
           s_loc[v][3] * z3 + s_loc[v][4] * z4) * sc - ln;
  }
  s_red[n * 16 + vb] = lp;
  __syncthreads();

  // ---- phase 3: TDM store of the samples tile + WMMA log-prob reduction ----
  if (tid < 32) {               // wave-uniform branch -> EXEC all ones
#if HAVE_TDM
    {   // fire the 20KB LDS -> HBM DMA first; it overlaps with the WMMAs below
      unsigned long long g =
          (unsigned long long)(uintptr_t)(void*)(out_samples + (size_t)b * TILE_F);
      v4u g0; v8i g1;
      tdm_desc_1d(g, (unsigned)(uintptr_t)(void*)&s_out[0], TILE_F, g0, g1);
      v4i gz = { 0, 0, 0, 0 };
#if defined(__clang_major__) && __clang_major__ >= 23
      v8i gz8 = { 0, 0, 0, 0, 0, 0, 0, 0 };
      __builtin_amdgcn_tensor_store_from_lds(g0, g1, gz, gz, gz8, 0);
#else
      __builtin_amdgcn_tensor_store_from_lds(g0, g1, gz, gz, 0);
#endif
    }
#endif
    // A[16x4] rows = sample n, cols = 4 of 16 vlane partials; B = ones =>
    // D[:,0] = row sums in exact f32. 4 chained WMMAs reduce all V=64.
    const int row = tid & 15;
    const int kk  = (tid >> 4) * 2;   // lanes 0-15: K=0,1 ; lanes 16-31: K=2,3
    v2f ones = { 1.0f, 1.0f };
    v8f acc  = { 0.f, 0.f, 0.f, 0.f, 0.f, 0.f, 0.f, 0.f };
#pragma unroll
    for (int j = 0; j < 4; ++j) {
      v2f a;
      a.x = s_red[row * 16 + 4 * j + kk];
      a.y = s_red[row * 16 + 4 * j + kk + 1];
      acc = __builtin_amdgcn_wmma_f32_16x16x4_f32(
          false, a, false, ones, (short)0, acc, false, false);
    }
    // D layout: lane 0 VGPR r -> (M=r, N=0); lane 16 VGPR r -> (M=8+r, N=0)
    const size_t lb = (size_t)b * N_SAMP;
    if (tid == 0) {
#pragma unroll
      for (int r2 = 0; r2 < 8; ++r2) out_logp[lb + r2] = acc[r2];
    } else if (tid == 16) {
#pragma unroll
      for (int r2 = 0; r2 < 8; ++r2) out_logp[lb + 8 + r2] = acc[r2];
    }
#if HAVE_TDM
    __builtin_amdgcn_s_wait_tensorcnt(0);   // (s_endpgm would wait-idle anyway)
#endif
  }

#if !HAVE_TDM
  // Fallback output path: coalesced 16B vector stores of the staged tile.
  {
    const float4* src = (const float4*)s_out;
    float4* dst = (float4*)(out_samples + (size_t)b * TILE_F);
    for (int i = tid; i < TILE_F / 4; i += 256) dst[i] = src[i];
  }
#endif
}

extern "C" void kernel_launch(void* const* d_in, const int* in_sizes, int n_in,
                              void* d_out, int out_size, void* d_ws, size_t ws_size,
                              hipStream_t stream) {
  (void)n_in; (void)d_ws; (void)ws_size; (void)out_size;
  const float* kmu = (const float*)d_in[0];          // [B, 64, 5] f32
  // d_in[1] = n_samples (==16, baked into the launch geometry)
  const int B = in_sizes[0] / ZROW;
  float* out_samples = (float*)d_out;                                   // [B,16,320]
  float* out_logp    = out_samples + (size_t)B * TILE_F;                // [B,16]
  vmf_product_posterior_kernel<<<B, 256, 0, stream>>>(kmu, out_samples, out_logp);
}